// SparseEmbedding_23261542875244
// MI455X (gfx1250) — compile-verified
//
#include <hip/hip_runtime.h>
#include <hip/hip_bf16.h>

// Sparse embedding gather for MI455X (gfx1250, wave32).
//
// out[row, :] = weight[indices[row], :]   row in [0, 204800), DIM = 256 fp32
//
// Pure HBM-bandwidth problem: ~420 MB total traffic -> ~18 us floor at
// 23.3 TB/s. Strategy: each wave32 handles TWO consecutive 1KB rows per
// iteration (4x b128 load + 4x b128 NT store in flight), indices fetched
// as one scalar int2 (uniform address -> SMEM path), and the NEXT pair of
// rows prefetched with global_prefetch_b8 to hide the dependent
// index->row load latency. Weight loads stay RT so the ~184 MB unique
// gathered set can live in the 192 MB L2 and serve duplicate indices;
// output stores are NT so the write-once stream doesn't evict it.

typedef float v4f __attribute__((ext_vector_type(4)));
typedef int   v2i __attribute__((ext_vector_type(2)));

#define EMB_DIM 256  // floats per row; 1024 bytes

__global__ __launch_bounds__(256) void SparseEmbedding_gather_kernel(
    const int*   __restrict__ indices,
    const float* __restrict__ weight,
    float*       __restrict__ out,
    int nrows)
{
    const int lane          = threadIdx.x & 31;   // lane within wave32
    const int waveInBlock   = threadIdx.x >> 5;
    const int wavesPerBlock = blockDim.x >> 5;
    const int waveStride    = gridDim.x * wavesPerBlock;   // waves in grid
    const int gwave         = blockIdx.x * wavesPerBlock + waveInBlock;

    const int rowStride = 2 * waveStride;

    for (int row = 2 * gwave; row < nrows; row += rowStride) {
        // Uniform (readfirstlane'd) address => compiler can use the scalar
        // SMEM path for the index pair; one load covers both rows.
        const int urow = __builtin_amdgcn_readfirstlane(row);
        v2i epair = *(const v2i*)(indices + urow);
        const int e0 = __builtin_amdgcn_readfirstlane(epair.x);
        const int e1 = __builtin_amdgcn_readfirstlane(epair.y);

        const v4f* __restrict__ src0 = (const v4f*)(weight + (size_t)e0 * EMB_DIM);
        const v4f* __restrict__ src1 = (const v4f*)(weight + (size_t)e1 * EMB_DIM);
        v4f*       __restrict__ dst0 = (v4f*)(out + (size_t)urow * EMB_DIM);
        v4f*       __restrict__ dst1 = (v4f*)(out + (size_t)(urow + 1) * EMB_DIM);

        // Prefetch next iteration's pair of rows (gfx1250 global_prefetch_b8).
        // 32 lanes x 32B stride cover each full 1KB row (8 x 128B lines).
        const int nrow = urow + rowStride;
        if (nrow + 1 < nrows) {
            v2i npair = *(const v2i*)(indices + nrow);
            const int en0 = __builtin_amdgcn_readfirstlane(npair.x);
            const int en1 = __builtin_amdgcn_readfirstlane(npair.y);
            const char* pf0 = (const char*)(weight + (size_t)en0 * EMB_DIM);
            const char* pf1 = (const char*)(weight + (size_t)en1 * EMB_DIM);
            __builtin_prefetch(pf0 + lane * 32, /*rw=*/0, /*locality=*/3);
            __builtin_prefetch(pf1 + lane * 32, /*rw=*/0, /*locality=*/3);
        }

        // Row = 256 floats: lane covers float4 slots {lane, lane+32}.
        // Issue all 4 loads before the stores so 4 b128 loads are in flight.
        v4f a0 = src0[lane];
        v4f b0 = src0[lane + 32];
        bool has1 = (urow + 1) < nrows;
        v4f a1, b1;
        if (has1) {
            a1 = src1[lane];
            b1 = src1[lane + 32];
        }

        // Write-once output stream: non-temporal so it doesn't thrash L2.
        __builtin_nontemporal_store(a0, dst0 + lane);
        __builtin_nontemporal_store(b0, dst0 + lane + 32);
        if (has1) {
            __builtin_nontemporal_store(a1, dst1 + lane);
            __builtin_nontemporal_store(b1, dst1 + lane + 32);
        }
    }
}

extern "C" void kernel_launch(void* const* d_in, const int* in_sizes, int n_in,
                              void* d_out, int out_size, void* d_ws, size_t ws_size,
                              hipStream_t stream) {
    const int*   indices = (const int*)  d_in[0];   // (4096, 50) int32
    const float* weight  = (const float*)d_in[1];   // (1e6, 256) float32
    float*       out     = (float*)d_out;           // (4096, 50, 256) float32

    const int nrows = in_sizes[0];                  // 4096*50 = 204800 rows

    // 2048 blocks x 8 waves = 16384 waves; 2 rows/wave/iter, ~6 iterations
    // per wave keeps the prefetch pipeline loop-carried and HBM saturated.
    const int threads = 256;
    const int blocks  = 2048;
    SparseEmbedding_gather_kernel<<<blocks, threads, 0, stream>>>(indices, weight, out, nrows);
}